// PGIN_81784767250527
// MI455X (gfx1250) — compile-verified
//
#include <hip/hip_runtime.h>
#include <hip/hip_bf16.h>
#include <math.h>

// GIN forward on MI455X (gfx1250).
//   N=100000 nodes, E=1600000 edges, F=64, 2F=128, C=10, L=4, G=512
// Aggregation is the roofline-dominant phase (L2-resident atomics); the MLP
// GEMMs (13 GFLOP total) run on full-precision V_WMMA_F32_16X16X4_F32 so we
// keep fp32 accuracy at negligible cost.

typedef __attribute__((ext_vector_type(2))) float v2f;
typedef __attribute__((ext_vector_type(8))) float v8f;

#define GIN_N 100000
#define GIN_E 1600000
#define GIN_F 64
#define GIN_F2 128
#define GIN_C 10
#define GIN_L 4
#define GIN_G 512

// ---------------------------------------------------------------- h = x copy
__global__ void gin_init_h(float* __restrict__ h, const float* __restrict__ x,
                           int n4) {
  int i = blockIdx.x * blockDim.x + threadIdx.x;
  if (i < n4) {
    reinterpret_cast<float4*>(h)[i] = reinterpret_cast<const float4*>(x)[i];
  }
}

// ------------------------------------------------- edge scatter: h[dst]+=x[src]
// thread t -> edge e = t/16, 16-byte feature chunk c = t%16 (coalesced).
__global__ void gin_aggregate(float* __restrict__ h, const float* __restrict__ x,
                              const int* __restrict__ src,
                              const int* __restrict__ dst, int nE) {
  int t = blockIdx.x * blockDim.x + threadIdx.x;
  int e = t >> 4;
  if (e >= nE) return;
  int c = (t & 15) << 2;
  int s = src[e];
  int d = dst[e];
  float4 v = *reinterpret_cast<const float4*>(x + (size_t)s * GIN_F + c);
  float* p = h + (size_t)d * GIN_F + c;
  atomicAdd(p + 0, v.x);
  atomicAdd(p + 1, v.y);
  atomicAdd(p + 2, v.z);
  atomicAdd(p + 3, v.w);
}

// ---------------------------------------------- fused MLP: relu(relu(H@W1)@W2)
// One wave32 per 16-row tile; 4 waves (128 threads) per block.
// f32 WMMA 16x16x4 fragment layout (ISA 7.12.2, 32-bit A 16x4 & symmetric B):
//   lane L: m/n = L&15, K pair = k0 + 2*(L>>4) + {0,1}
//   C/D:    VGPR r -> (M = r + 8*(L>>4), N = L&15)
#define HS_STRIDE 66   // 16x64 H tile, +2 pad
#define TS_STRIDE 130  // 16x128 T tile, +2 pad

__global__ __launch_bounds__(128) void gin_mlp(
    const float* __restrict__ h, const float* __restrict__ W1,
    const float* __restrict__ W2, float* __restrict__ xout, int nTiles) {
  __shared__ float Hs[4][16 * HS_STRIDE];
  __shared__ float Ts[4][16 * TS_STRIDE];

  const int lane = threadIdx.x & 31;
  const int wave = threadIdx.x >> 5;
  const int tile = blockIdx.x * 4 + wave;
  if (tile >= nTiles) return;  // wave-uniform exit: EXEC stays all-1s for WMMA

  float* hs = Hs[wave];
  float* ts = Ts[wave];
  const size_t rowBase = (size_t)tile * 16;

  // Stage 16x64 H tile into LDS (b128 coalesced).
  const float* hsrc = h + rowBase * GIN_F;
  #pragma unroll
  for (int i = 0; i < 8; ++i) {
    int f = (lane + 32 * i) << 2;  // 0..1020
    int r = f >> 6;
    int c = f & 63;
    float4 v = *reinterpret_cast<const float4*>(hsrc + f);
    float* p = hs + r * HS_STRIDE + c;
    p[0] = v.x; p[1] = v.y; p[2] = v.z; p[3] = v.w;
  }
  // LDS ops from one wave are in-order (DScnt); no cross-wave sharing -> no barrier.

  const int m = lane & 15;
  const int koff = (lane >> 4) << 1;  // 0 or 2

  // GEMM1: T[16x128] = relu(H[16x64] @ W1[64x128])
  #pragma unroll
  for (int nt = 0; nt < 8; ++nt) {
    const int c = nt * 16 + m;
    v8f acc = {};
    #pragma unroll
    for (int k0 = 0; k0 < GIN_F; k0 += 4) {
      const int kk = k0 + koff;
      v2f a, b;
      a[0] = hs[m * HS_STRIDE + kk];
      a[1] = hs[m * HS_STRIDE + kk + 1];
      b[0] = W1[kk * GIN_F2 + c];
      b[1] = W1[(kk + 1) * GIN_F2 + c];
      acc = __builtin_amdgcn_wmma_f32_16x16x4_f32(false, a, false, b,
                                                  (short)0, acc, false, false);
    }
    #pragma unroll
    for (int r = 0; r < 8; ++r) {
      float v = acc[r];
      v = v > 0.f ? v : 0.f;                       // ReLU #1
      ts[(r + (koff << 2)) * TS_STRIDE + c] = v;   // M = r + 8*(lane>>4)
    }
  }

  // GEMM2: X[16x64] = relu(T[16x128] @ W2[128x64])
  #pragma unroll
  for (int nt = 0; nt < 4; ++nt) {
    const int c = nt * 16 + m;
    v8f acc = {};
    #pragma unroll
    for (int k0 = 0; k0 < GIN_F2; k0 += 4) {
      const int kk = k0 + koff;
      v2f a, b;
      a[0] = ts[m * TS_STRIDE + kk];
      a[1] = ts[m * TS_STRIDE + kk + 1];
      b[0] = W2[kk * GIN_F + c];
      b[1] = W2[(kk + 1) * GIN_F + c];
      acc = __builtin_amdgcn_wmma_f32_16x16x4_f32(false, a, false, b,
                                                  (short)0, acc, false, false);
    }
    float* xo = xout + (rowBase + (size_t)(koff << 2)) * GIN_F + c;
    #pragma unroll
    for (int r = 0; r < 8; ++r) {
      float v = acc[r];
      v = v > 0.f ? v : 0.f;  // ReLU #2
      xo[(size_t)r * GIN_F] = v;
    }
  }
}

// ------------------------------------------------------------------- pooling
__global__ void gin_zero(float* __restrict__ p, int n4) {
  int i = blockIdx.x * blockDim.x + threadIdx.x;
  if (i < n4) reinterpret_cast<float4*>(p)[i] = make_float4(0.f, 0.f, 0.f, 0.f);
}

__global__ void gin_pool(float* __restrict__ pooled, const float* __restrict__ x,
                         const int* __restrict__ batch, int nN) {
  int t = blockIdx.x * blockDim.x + threadIdx.x;
  int n = t >> 4;
  if (n >= nN) return;
  int c = (t & 15) << 2;
  int b = batch[n];
  float4 v = *reinterpret_cast<const float4*>(x + (size_t)n * GIN_F + c);
  float* p = pooled + (size_t)b * GIN_F + c;
  atomicAdd(p + 0, v.x);
  atomicAdd(p + 1, v.y);
  atomicAdd(p + 2, v.z);
  atomicAdd(p + 3, v.w);
}

// ------------------------------------------------- FC head + exact log_softmax
__global__ void gin_head(const float* __restrict__ pooled,
                         const float* __restrict__ fcW,
                         const float* __restrict__ fcb,
                         float* __restrict__ out, int nG) {
  int g = blockIdx.x * blockDim.x + threadIdx.x;
  if (g >= nG) return;
  float acc[GIN_C];
  #pragma unroll
  for (int j = 0; j < GIN_C; ++j) acc[j] = fcb[j];
  const float* pr = pooled + (size_t)g * GIN_F;
  for (int k = 0; k < GIN_F; ++k) {
    float p = pr[k];
    #pragma unroll
    for (int j = 0; j < GIN_C; ++j) acc[j] += p * fcW[k * GIN_C + j];
  }
  float mx = acc[0];
  #pragma unroll
  for (int j = 1; j < GIN_C; ++j) mx = fmaxf(mx, acc[j]);
  float s = 0.f;
  #pragma unroll
  for (int j = 0; j < GIN_C; ++j) s += expf(acc[j] - mx);
  float ls = logf(s);
  #pragma unroll
  for (int j = 0; j < GIN_C; ++j) out[(size_t)g * GIN_C + j] = acc[j] - mx - ls;
}

// ---------------------------------------------------------------------- launch
extern "C" void kernel_launch(void* const* d_in, const int* in_sizes, int n_in,
                              void* d_out, int out_size, void* d_ws,
                              size_t ws_size, hipStream_t stream) {
  // setup_inputs order: x, W1, W2, fcW, fcb, edge_index, batch
  const float* x_in = (const float*)d_in[0];
  const float* W1   = (const float*)d_in[1];  // [L,64,128]
  const float* W2   = (const float*)d_in[2];  // [L,128,64]
  const float* fcW  = (const float*)d_in[3];  // [64,10]
  const float* fcb  = (const float*)d_in[4];  // [10]
  const int* eidx   = (const int*)d_in[5];    // [2,E]
  const int* batch  = (const int*)d_in[6];    // [N]
  float* out        = (float*)d_out;          // [G,C]

  const int* src = eidx;
  const int* dst = eidx + GIN_E;

  float* ws     = (float*)d_ws;
  float* x_buf  = ws;                              // N*F
  float* h      = ws + (size_t)GIN_N * GIN_F;      // N*F
  float* pooled = ws + (size_t)2 * GIN_N * GIN_F;  // G*F

  const int n4    = GIN_N * GIN_F / 4;     // 1.6M float4
  const int nTile = GIN_N / 16;            // 6250 (exact)

  for (int l = 0; l < GIN_L; ++l) {
    const float* xin = (l == 0) ? x_in : x_buf;
    gin_init_h<<<(n4 + 255) / 256, 256, 0, stream>>>(h, xin, n4);
    gin_aggregate<<<((size_t)GIN_E * 16 + 255) / 256, 256, 0, stream>>>(
        h, xin, src, dst, GIN_E);
    gin_mlp<<<(nTile + 3) / 4, 128, 0, stream>>>(
        h, W1 + (size_t)l * GIN_F * GIN_F2, W2 + (size_t)l * GIN_F2 * GIN_F,
        x_buf, nTile);
  }

  const int g4 = GIN_G * GIN_F / 4;
  gin_zero<<<(g4 + 255) / 256, 256, 0, stream>>>(pooled, g4);
  gin_pool<<<((size_t)GIN_N * 16 + 255) / 256, 256, 0, stream>>>(
      pooled, x_buf, batch, GIN_N);
  gin_head<<<(GIN_G + 255) / 256, 256, 0, stream>>>(pooled, fcW, fcb, out,
                                                    GIN_G);
}